// HierarchicalGAT_12017318494744
// MI455X (gfx1250) — compile-verified
//
#include <hip/hip_runtime.h>
#include <hip/hip_bf16.h>
#include <math.h>

typedef float v2f __attribute__((ext_vector_type(2)));
typedef float v8f __attribute__((ext_vector_type(8)));

#define BNEPS 1e-5f
#define NEGSLOPE 0.2f

// ---------------------------------------------------------------- utilities
__global__ void fill_kernel(float* __restrict__ p, float v, long long n) {
  long long i = blockIdx.x * (long long)blockDim.x + threadIdx.x;
  if (i < n) p[i] = v;
}

__device__ __forceinline__ void edge_nodes(const int* __restrict__ ei, long long E,
                                           long long e, int& s, int& d) {
  if (e < E) { s = ei[e]; d = ei[E + e]; }
  else       { s = d = (int)(e - E); }            // PyG-style self loop
}

// ------------------------------------------------- fp32 WMMA GEMM  H = X @ W
// One wave32 per 16x16 output tile. X:[nrows,K] row-major, W:[K,MC] row-major.
template <int K, int MC>
__global__ __launch_bounds__(256) void gemm_wmma_f32(const float* __restrict__ X,
                                                     const float* __restrict__ W,
                                                     float* __restrict__ H,
                                                     int nrows) {
  int wid  = (int)((blockIdx.x * blockDim.x + threadIdx.x) >> 5);
  int lane = threadIdx.x & 31;
  const int ntc = MC / 16;
  int tm = wid / ntc;
  int tn = wid - tm * ntc;
  if (tm * 16 >= nrows) return;                    // whole-wave uniform exit

  int m  = lane & 15;                              // M (for A) / N (for B,D)
  int kh = (lane >> 4) << 1;                       // K sub-offset: 0 or 2
  const float* xrow = X + (long long)(tm * 16 + m) * K;
  int col = tn * 16 + m;

  v8f acc = {};
#pragma unroll
  for (int k0 = 0; k0 < K; k0 += 4) {
    v2f a, b;
    a.x = xrow[k0 + kh];
    a.y = xrow[k0 + kh + 1];
    b.x = W[(long long)(k0 + kh) * MC + col];
    b.y = W[(long long)(k0 + kh + 1) * MC + col];
    acc = __builtin_amdgcn_wmma_f32_16x16x4_f32(false, a, false, b,
                                                (short)0, acc, false, false);
  }
  int rbase = (lane >> 4) * 8;                     // D layout: VGPR r -> M=r / M=r+8
#pragma unroll
  for (int r = 0; r < 8; ++r)
    H[(long long)(tm * 16 + rbase + r) * MC + col] = acc[r];
}

// ------------------------------------------- per-node attention projections
// a_s[n,h] = <h[n,h,:], att_src[h,:]> ; a_d likewise. C = 64.
__global__ void attn_scores(const float* __restrict__ Hf,
                            const float* __restrict__ aws,
                            const float* __restrict__ awd,
                            float* __restrict__ as_, float* __restrict__ ad_,
                            int n, int heads) {
  long long i = blockIdx.x * (long long)blockDim.x + threadIdx.x;
  if (i >= (long long)n * heads) return;
  int node = (int)(i / heads), h = (int)(i % heads);
  const float* hp = Hf + ((long long)node * heads + h) * 64;
  const float* ws = aws + h * 64;
  const float* wd = awd + h * 64;
  float s = 0.f, d = 0.f;
#pragma unroll 8
  for (int c = 0; c < 64; ++c) { float v = hp[c]; s += v * ws[c]; d += v * wd[c]; }
  as_[i] = s; ad_[i] = d;
}

// ------------------------------------------------- edge pass 1: lrelu + max
__global__ void edge_lrelu_max(const int* __restrict__ ei, long long E, int n, int Hh,
                               const float* __restrict__ as_, const float* __restrict__ ad_,
                               float* __restrict__ ev, float* __restrict__ mx) {
  long long nE = E + n;
  long long i = blockIdx.x * (long long)blockDim.x + threadIdx.x;
  if (i >= nE * Hh) return;
  long long e = i / Hh; int h = (int)(i % Hh);
  int s, d; edge_nodes(ei, E, e, s, d);
  float v = as_[(long long)s * Hh + h] + ad_[(long long)d * Hh + h];
  v = v > 0.f ? v : NEGSLOPE * v;
  ev[i] = v;
  atomicMax(&mx[(long long)d * Hh + h], v);        // global_atomic_max_num_f32
}

// ------------------------------------------------- edge pass 2: exp + sum
__global__ void edge_exp_sum(const int* __restrict__ ei, long long E, int n, int Hh,
                             const float* __restrict__ mx,
                             float* __restrict__ ev, float* __restrict__ den) {
  long long nE = E + n;
  long long i = blockIdx.x * (long long)blockDim.x + threadIdx.x;
  if (i >= nE * Hh) return;
  long long e = i / Hh; int h = (int)(i % Hh);
  int s, d; edge_nodes(ei, E, e, s, d);
  (void)s;
  float t = expf(ev[i] - mx[(long long)d * Hh + h]);
  ev[i] = t;
  atomicAdd(&den[(long long)d * Hh + h], t);
}

// ------------------------- edge pass 3: wave-per-edge weighted scatter-add
template <int Hh, int C>
__global__ __launch_bounds__(256) void edge_aggregate(const int* __restrict__ ei,
                                                      long long E, int n,
                                                      const float* __restrict__ feat,
                                                      const float* __restrict__ ev,
                                                      const float* __restrict__ den,
                                                      float* __restrict__ out) {
  const int HC = Hh * C;
  long long wave = (blockIdx.x * (long long)blockDim.x + threadIdx.x) >> 5;
  int lane = threadIdx.x & 31;
  long long nE = E + n;
  if (wave >= nE) return;
  int s, d; edge_nodes(ei, E, wave, s, d);
  const float* src = feat + (long long)s * HC;
  float* dst = out + (long long)d * HC;
#pragma unroll
  for (int c0 = lane * 4; c0 < HC; c0 += 128) {    // float4 lanes, coalesced
    int h = c0 / C;
    float w = ev[wave * Hh + h] / (den[(long long)d * Hh + h] + 1e-16f);
    float4 v = *reinterpret_cast<const float4*>(src + c0);
    atomicAdd(dst + c0 + 0, v.x * w);
    atomicAdd(dst + c0 + 1, v.y * w);
    atomicAdd(dst + c0 + 2, v.z * w);
    atomicAdd(dst + c0 + 3, v.w * w);
  }
}

// ------------------------------------ bias + BN(eval) + ELU (layer1 epilog)
__global__ void bias_bn_elu(float* __restrict__ x, const float* __restrict__ b,
                            const float* __restrict__ g, const float* __restrict__ be,
                            const float* __restrict__ m, const float* __restrict__ v,
                            long long total, int C) {
  long long i = blockIdx.x * (long long)blockDim.x + threadIdx.x;
  if (i >= total) return;
  int c = (int)(i % C);
  float t = x[i] + b[c];
  t = (t - m[c]) * (g[c] * rsqrtf(v[c] + BNEPS)) + be[c];
  x[i] = t > 0.f ? t : (expf(t) - 1.f);
}

// -------------- head-mean + bias + BN2 + classifier (HID=64 -> NC=2 logits)
__global__ void finalize(const float* __restrict__ out2, const float* __restrict__ b2,
                         const float* __restrict__ g2, const float* __restrict__ be2,
                         const float* __restrict__ m2, const float* __restrict__ v2,
                         const float* __restrict__ wc, const float* __restrict__ bc,
                         float* __restrict__ logits, int n) {
  int node = blockIdx.x * blockDim.x + threadIdx.x;
  if (node >= n) return;
  const float* o = out2 + (long long)node * 128;   // [2 heads, 64]
  float l0 = bc[0], l1 = bc[1];
#pragma unroll 8
  for (int c = 0; c < 64; ++c) {
    float t = 0.5f * (o[c] + o[64 + c]) + b2[c];
    t = (t - m2[c]) * (g2[c] * rsqrtf(v2[c] + BNEPS)) + be2[c];
    l0 += t * wc[c * 2 + 0];
    l1 += t * wc[c * 2 + 1];
  }
  logits[node * 2 + 0] = l0;
  logits[node * 2 + 1] = l1;
}

// ---------------------------------------------------------------------------
extern "C" void kernel_launch(void* const* d_in, const int* in_sizes, int n_in,
                              void* d_out, int out_size, void* d_ws, size_t ws_size,
                              hipStream_t stream) {
  (void)n_in; (void)out_size; (void)ws_size;
  const float* x    = (const float*)d_in[0];
  const int*   ei   = (const int*)d_in[1];
  const float* w1   = (const float*)d_in[2];
  const float* as1w = (const float*)d_in[3];
  const float* ad1w = (const float*)d_in[4];
  const float* b1   = (const float*)d_in[5];
  const float* g1   = (const float*)d_in[6];
  const float* be1  = (const float*)d_in[7];
  const float* m1   = (const float*)d_in[8];
  const float* v1   = (const float*)d_in[9];
  const float* w2   = (const float*)d_in[10];
  const float* as2w = (const float*)d_in[11];
  const float* ad2w = (const float*)d_in[12];
  const float* b2   = (const float*)d_in[13];
  const float* g2   = (const float*)d_in[14];
  const float* be2  = (const float*)d_in[15];
  const float* m2   = (const float*)d_in[16];
  const float* v2   = (const float*)d_in[17];
  const float* wc   = (const float*)d_in[18];
  const float* bc   = (const float*)d_in[19];

  const int       n  = in_sizes[0] / 128;          // 100000 (multiple of 16)
  const long long E  = in_sizes[1] / 2;            // 1.6M
  const long long nE = E + n;

  // ---- workspace carve (floats) -------------------------------------------
  float* ws   = (float*)d_ws;
  float* h1   = ws;                                 // n*256
  float* out1 = h1   + (long long)n * 256;          // n*256
  float* asb  = out1 + (long long)n * 256;          // n*4
  float* adb  = asb  + (long long)n * 4;            // n*4
  float* mx   = adb  + (long long)n * 4;            // n*4
  float* den  = mx   + (long long)n * 4;            // n*4
  float* ev   = den  + (long long)n * 4;            // nE*4
  float* h2   = h1;                                 // reuse: n*128
  float* out2 = h1 + (long long)n * 128;            // reuse: n*128

  const int B = 256;
  auto blk1 = [](long long t) { return (unsigned)((t + 255) / 256); };

  // ======================= layer 1: GATConv(128 -> 4x64, concat) ============
  {
    long long waves = ((long long)(n + 15) / 16) * (256 / 16);
    gemm_wmma_f32<128, 256><<<blk1(waves * 32), B, 0, stream>>>(x, w1, h1, n);
  }
  attn_scores<<<blk1((long long)n * 4), B, 0, stream>>>(h1, as1w, ad1w, asb, adb, n, 4);

  fill_kernel<<<blk1((long long)n * 4), B, 0, stream>>>(mx, -3.0e38f, (long long)n * 4);
  fill_kernel<<<blk1((long long)n * 4), B, 0, stream>>>(den, 0.f, (long long)n * 4);
  fill_kernel<<<blk1((long long)n * 256), B, 0, stream>>>(out1, 0.f, (long long)n * 256);

  edge_lrelu_max<<<blk1(nE * 4), B, 0, stream>>>(ei, E, n, 4, asb, adb, ev, mx);
  edge_exp_sum  <<<blk1(nE * 4), B, 0, stream>>>(ei, E, n, 4, mx, ev, den);
  edge_aggregate<4, 64><<<blk1(nE * 32), B, 0, stream>>>(ei, E, n, h1, ev, den, out1);

  bias_bn_elu<<<blk1((long long)n * 256), B, 0, stream>>>(out1, b1, g1, be1, m1, v1,
                                                          (long long)n * 256, 256);

  // ======================= layer 2: GATConv(256 -> mean of 2x64) ============
  {
    long long waves = ((long long)(n + 15) / 16) * (128 / 16);
    gemm_wmma_f32<256, 128><<<blk1(waves * 32), B, 0, stream>>>(out1, w2, h2, n);
  }
  attn_scores<<<blk1((long long)n * 2), B, 0, stream>>>(h2, as2w, ad2w, asb, adb, n, 2);

  fill_kernel<<<blk1((long long)n * 2), B, 0, stream>>>(mx, -3.0e38f, (long long)n * 2);
  fill_kernel<<<blk1((long long)n * 2), B, 0, stream>>>(den, 0.f, (long long)n * 2);
  fill_kernel<<<blk1((long long)n * 128), B, 0, stream>>>(out2, 0.f, (long long)n * 128);

  edge_lrelu_max<<<blk1(nE * 2), B, 0, stream>>>(ei, E, n, 2, asb, adb, ev, mx);
  edge_exp_sum  <<<blk1(nE * 2), B, 0, stream>>>(ei, E, n, 2, mx, ev, den);
  edge_aggregate<2, 64><<<blk1(nE * 32), B, 0, stream>>>(ei, E, n, h2, ev, den, out2);

  // ======================= head-mean + BN2 + classifier =====================
  finalize<<<blk1(n), B, 0, stream>>>(out2, b2, g2, be2, m2, v2, wc, bc,
                                      (float*)d_out, n);
}